// GravityHypothesisTester_63376537419914
// MI455X (gfx1250) — compile-verified
//
#include <hip/hip_runtime.h>
#include <math.h>

typedef unsigned long long ull;
typedef __attribute__((ext_vector_type(2))) float v2f;
typedef __attribute__((ext_vector_type(8))) float v8f;

#define EPSF 1e-6f
#define CHI2F 9.0f
#define DSCALE 3.0f

// ---------- helpers ----------
__device__ __forceinline__ unsigned ordf(float f) {
    unsigned u = __float_as_uint(f);
    return (u & 0x80000000u) ? ~u : (u | 0x80000000u);
}
__device__ __forceinline__ float unordf(unsigned u) {
    return (u & 0x80000000u) ? __uint_as_float(u ^ 0x80000000u)
                             : __uint_as_float(~u);
}
__device__ __forceinline__ ull umin64(ull a, ull b) { return a < b ? a : b; }
__device__ __forceinline__ ull shflxor64(ull v, int mask) {
    unsigned lo = (unsigned)__shfl_xor((int)(unsigned)(v & 0xffffffffull), mask, 32);
    unsigned hi = (unsigned)__shfl_xor((int)(unsigned)(v >> 32), mask, 32);
    return ((ull)hi << 32) | (ull)lo;
}

// ---------- kernel 1: rotation + centering prep ----------
__global__ void k_prep(const float* __restrict__ src, const float* __restrict__ tgt,
                       const float* __restrict__ src_n,
                       const float* __restrict__ g_p, const float* __restrict__ g_q,
                       float* __restrict__ SI, float* __restrict__ SNR,
                       float* __restrict__ XX, float* __restrict__ YY,
                       ull* __restrict__ COLK, int N) {
    int b = blockIdx.x;
    int tid = threadIdx.x;
    int nthr = blockDim.x;

    // --- gravity_align, computed redundantly per thread (cheap) ---
    float gpx = g_p[b * 3 + 0], gpy = g_p[b * 3 + 1], gpz = g_p[b * 3 + 2];
    float gqx = g_q[b * 3 + 0], gqy = g_q[b * 3 + 1], gqz = g_q[b * 3 + 2];
    float np = fmaxf(sqrtf(gpx * gpx + gpy * gpy + gpz * gpz), EPSF);
    float nq = fmaxf(sqrtf(gqx * gqx + gqy * gqy + gqz * gqz), EPSF);
    float ux = gpx / np, uy = gpy / np, uz = gpz / np;
    float vx = gqx / nq, vy = gqy / nq, vz = gqz / nq;
    float ax = uy * vz - uz * vy;
    float ay = uz * vx - ux * vz;
    float az = ux * vy - uy * vx;
    float an = sqrtf(ax * ax + ay * ay + az * az);
    float duv = fminf(fmaxf(ux * vx + uy * vy + uz * vz, -1.0f), 1.0f);
    bool parallel = an < 1e-6f;
    float kx = ax / (an + EPSF), ky = ay / (an + EPSF), kz = az / (an + EPSF);
    float theta = acosf(duv);
    float st = sinf(theta), ct = cosf(theta);
    float K[3][3] = {{0.f, -kz, ky}, {kz, 0.f, -kx}, {-ky, kx, 0.f}};
    float K2[3][3];
    for (int i = 0; i < 3; ++i)
        for (int j = 0; j < 3; ++j)
            K2[i][j] = K[i][0] * K[0][j] + K[i][1] * K[1][j] + K[i][2] * K[2][j];
    float R[3][3];
    for (int i = 0; i < 3; ++i)
        for (int j = 0; j < 3; ++j)
            R[i][j] = (i == j ? 1.0f : 0.0f) + st * K[i][j] + (1.0f - ct) * K2[i][j];
    // antiparallel branch
    float bx = (fabsf(ux) < 0.9f) ? 1.0f : 0.0f;
    float by = 1.0f - bx;
    float cx2 = uy * 0.0f - uz * by;
    float cy2 = uz * bx - ux * 0.0f;
    float cz2 = ux * by - uy * bx;
    float n2 = fmaxf(sqrtf(cx2 * cx2 + cy2 * cy2 + cz2 * cz2), EPSF);
    float a2x = cx2 / n2, a2y = cy2 / n2, a2z = cz2 / n2;
    float Ks[3][3] = {{0.f, -a2z, a2y}, {a2z, 0.f, -a2x}, {-a2y, a2x, 0.f}};
    float Ks2[3][3];
    for (int i = 0; i < 3; ++i)
        for (int j = 0; j < 3; ++j)
            Ks2[i][j] = Ks[i][0] * Ks[0][j] + Ks[i][1] * Ks[1][j] + Ks[i][2] * Ks[2][j];
    bool antipar = parallel && (duv < 0.0f);
    bool parpos = parallel && (duv > 0.0f);
    for (int i = 0; i < 3; ++i)
        for (int j = 0; j < 3; ++j) {
            float ra = (i == j ? 1.0f : 0.0f) + 2.0f * Ks2[i][j];
            float rr = antipar ? ra : R[i][j];
            R[i][j] = parpos ? (i == j ? 1.0f : 0.0f) : rr;
        }

    const float* S0 = src + (size_t)b * 3 * N;
    const float* S1 = S0 + N;
    const float* S2 = S0 + 2 * N;
    const float* T0 = tgt + (size_t)b * 3 * N;
    const float* T1 = T0 + N;
    const float* T2 = T0 + 2 * N;
    const float* N0 = src_n + (size_t)b * 3 * N;
    const float* N1 = N0 + N;
    const float* N2 = N0 + 2 * N;

    // --- pass 1: sums of R*src and tgt ---
    float acc[6] = {0.f, 0.f, 0.f, 0.f, 0.f, 0.f};
    for (int n = tid; n < N; n += nthr) {
        float x = S0[n], y = S1[n], z = S2[n];
        acc[0] += R[0][0] * x + R[0][1] * y + R[0][2] * z;
        acc[1] += R[1][0] * x + R[1][1] * y + R[1][2] * z;
        acc[2] += R[2][0] * x + R[2][1] * y + R[2][2] * z;
        acc[3] += T0[n];
        acc[4] += T1[n];
        acc[5] += T2[n];
    }
    __shared__ float red[256];
    __shared__ float sums[6];
    for (int c = 0; c < 6; ++c) {
        red[tid] = acc[c];
        __syncthreads();
        for (int s = nthr >> 1; s > 0; s >>= 1) {
            if (tid < s) red[tid] += red[tid + s];
            __syncthreads();
        }
        if (tid == 0) sums[c] = red[0];
        __syncthreads();
    }
    float invN = 1.0f / (float)N;
    float tcx = (sums[3] - sums[0]) * invN;
    float tcy = (sums[4] - sums[1]) * invN;
    float tcz = (sums[5] - sums[2]) * invN;

    // --- pass 2: write src_init, src_n_rot, xx, yy; init column keys ---
    float* SI0 = SI + (size_t)b * 3 * N;
    float* SI1 = SI0 + N;
    float* SI2 = SI0 + 2 * N;
    float* SR0 = SNR + (size_t)b * 3 * N;
    float* SR1 = SR0 + N;
    float* SR2 = SR0 + 2 * N;
    for (int n = tid; n < N; n += nthr) {
        float x = S0[n], y = S1[n], z = S2[n];
        float rx = R[0][0] * x + R[0][1] * y + R[0][2] * z + tcx;
        float ry = R[1][0] * x + R[1][1] * y + R[1][2] * z + tcy;
        float rz = R[2][0] * x + R[2][1] * y + R[2][2] * z + tcz;
        SI0[n] = rx; SI1[n] = ry; SI2[n] = rz;
        XX[(size_t)b * N + n] = rx * rx + ry * ry + rz * rz;
        float tx = T0[n], ty = T1[n], tz = T2[n];
        YY[(size_t)b * N + n] = tx * tx + ty * ty + tz * tz;
        float nx = N0[n], ny = N1[n], nz = N2[n];
        SR0[n] = R[0][0] * nx + R[0][1] * ny + R[0][2] * nz;
        SR1[n] = R[1][0] * nx + R[1][1] * ny + R[1][2] * nz;
        SR2[n] = R[2][0] * nx + R[2][1] * ny + R[2][2] * nz;
        COLK[(size_t)b * N + n] = ~0ull;
    }
}

// ---------- kernel 2: WMMA pairwise distances + min/argmin keys ----------
__global__ void k_dist(const float* __restrict__ SI, const float* __restrict__ XX,
                       const float* __restrict__ tgt, const float* __restrict__ YY,
                       ull* __restrict__ ROWK, ull* __restrict__ COLK, int N) {
    int b = blockIdx.y;
    int lane = threadIdx.x & 31;
    int wave = threadIdx.x >> 5;
    int tile = blockIdx.x * (blockDim.x >> 5) + wave;
    int m0 = tile * 16;
    if (m0 >= N) return;

    const float* S0 = SI + (size_t)b * 3 * N;
    const float* S1 = S0 + N;
    const float* S2 = S0 + 2 * N;
    const float* T0 = tgt + (size_t)b * 3 * N;
    const float* T1 = T0 + N;
    const float* T2 = T0 + 2 * N;
    const float* xx = XX + (size_t)b * N;
    const float* yy = YY + (size_t)b * N;

    int half = lane >> 4;       // 0: K=0,1   1: K=2,3
    int l = lane & 15;
    int r = m0 + l;

    // A matrix 16x4 f32: row m = (-2sx, -2sy, -2sz, xx[m])
    v2f a;
    if (half == 0) { a.x = -2.0f * S0[r]; a.y = -2.0f * S1[r]; }
    else           { a.x = -2.0f * S2[r]; a.y = xx[r]; }

    ull rk[8];
#pragma unroll
    for (int j = 0; j < 8; ++j) rk[j] = ~0ull;

    for (int c0 = 0; c0 < N; c0 += 16) {
        // B matrix 4x16 f32: col n = (tx, ty, tz, 1)
        v2f bb;
        if (half == 0) { bb.x = T0[c0 + l]; bb.y = T1[c0 + l]; }
        else           { bb.x = T2[c0 + l]; bb.y = 1.0f; }
        // C = yy[n] broadcast (N index = lane&15 for every C VGPR)
        float cy = yy[c0 + l];
        v8f c;
#pragma unroll
        for (int j = 0; j < 8; ++j) c[j] = cy;

        v8f d = __builtin_amdgcn_wmma_f32_16x16x4_f32(
            false, a, false, bb, (short)0, c, false, false);

        ull ck = ~0ull;
#pragma unroll
        for (int j = 0; j < 8; ++j) {
            unsigned od = ordf(d[j]);
            int m = m0 + j + (half ? 8 : 0);
            int n = c0 + l;
            rk[j] = umin64(rk[j], ((ull)od << 32) | (unsigned)n);
            ck = umin64(ck, ((ull)od << 32) | (unsigned)m);
        }
        // combine the two half-waves that hold the same column, then atomic
        ck = umin64(ck, shflxor64(ck, 16));
        if (half == 0)
            atomicMin(&COLK[(size_t)b * N + c0 + l], ck);
    }

    // row reduce within each 16-lane half (each half holds distinct rows)
#pragma unroll
    for (int off = 8; off >= 1; off >>= 1) {
#pragma unroll
        for (int j = 0; j < 8; ++j)
            rk[j] = umin64(rk[j], shflxor64(rk[j], off));
    }
    if (l == 0) {
#pragma unroll
        for (int j = 0; j < 8; ++j)
            ROWK[(size_t)b * N + m0 + j + (half ? 8 : 0)] = rk[j];
    }
}

// ---------- kernel 3: nearest-neighbor distances ----------
__global__ void k_nn(const ull* __restrict__ ROWK, const ull* __restrict__ COLK,
                     float* __restrict__ NDP, float* __restrict__ NDQ, int total) {
    int i = blockIdx.x * blockDim.x + threadIdx.x;
    if (i >= total) return;
    float dp = unordf((unsigned)(ROWK[i] >> 32));
    NDP[i] = sqrtf(fmaxf(dp, 0.0f) + EPSF);
    float dq = unordf((unsigned)(COLK[i] >> 32));
    NDQ[i] = sqrtf(fmaxf(dq, 0.0f) + EPSF);
}

// ---------- kernel 4: exact lower-median via rank counting ----------
__global__ void k_median(const float* __restrict__ NDP, const float* __restrict__ NDQ,
                         float* __restrict__ TAU, int B, int N) {
    extern __shared__ float ls[];
    int bid = blockIdx.x;  // 0..2B-1 : first B = p, next B = q
    const float* arr = (bid < B) ? (NDP + (size_t)bid * N)
                                 : (NDQ + (size_t)(bid - B) * N);
    int tid = threadIdx.x, nthr = blockDim.x;
    for (int i = tid; i < N; i += nthr) ls[i] = arr[i];
    __syncthreads();
    int k = (N - 1) / 2;
    for (int i = tid; i < N; i += nthr) {
        float x = ls[i];
        int cl = 0, ce = 0;
        for (int j = 0; j < N; ++j) {
            float y = ls[j];
            cl += (y < x);
            ce += (y == x);
        }
        if (cl <= k && k < cl + ce) TAU[bid] = DSCALE * x;
    }
}

// ---------- kernel 5: incidence weights ----------
__global__ void k_weights(const float* __restrict__ SNR, const float* __restrict__ tgt_n,
                          const ull* __restrict__ ROWK, const ull* __restrict__ COLK,
                          const float* __restrict__ NDP, const float* __restrict__ NDQ,
                          const float* __restrict__ TAU,
                          const float* __restrict__ g_q, const float* __restrict__ k_p,
                          const float* __restrict__ k_q,
                          float* __restrict__ out, int B, int N) {
    int b = blockIdx.y;
    int n = blockIdx.x * blockDim.x + threadIdx.x;
    if (n >= N) return;
    float gx = g_q[b * 3 + 0], gy = g_q[b * 3 + 1], gz = g_q[b * 3 + 2];
    float ke = k_p[b] * k_q[b] / (k_p[b] + k_q[b] + EPSF);
    const float* SN0 = SNR + (size_t)b * 3 * N;
    const float* SN1 = SN0 + N;
    const float* SN2 = SN0 + 2 * N;
    const float* TN0 = tgt_n + (size_t)b * 3 * N;
    const float* TN1 = TN0 + N;
    const float* TN2 = TN0 + 2 * N;

    // w_p
    ull rk = ROWK[(size_t)b * N + n];
    int jp = (int)(unsigned)(rk & 0xffffffffull);
    float inc_p = SN0[n] * gx + SN1[n] * gy + SN2[n] * gz;
    float inc_p_ref = TN0[jp] * gx + TN1[jp] * gy + TN2[jp] * gz;
    float geom_p = (NDP[(size_t)b * N + n] <= TAU[b]) ? 1.0f : 0.0f;
    float dp = inc_p - inc_p_ref;
    float wp = (1.0f / (1.0f + expf(-(CHI2F - ke * dp * dp)))) * geom_p;
    out[(size_t)b * N + n] = wp;

    // w_q
    ull ck = COLK[(size_t)b * N + n];
    int jq = (int)(unsigned)(ck & 0xffffffffull);
    float inc_q = TN0[n] * gx + TN1[n] * gy + TN2[n] * gz;
    float inc_q_ref = SN0[jq] * gx + SN1[jq] * gy + SN2[jq] * gz;
    float geom_q = (NDQ[(size_t)b * N + n] <= TAU[B + b]) ? 1.0f : 0.0f;
    float dq = inc_q - inc_q_ref;
    float wq = (1.0f / (1.0f + expf(-(CHI2F - ke * dq * dq)))) * geom_q;
    out[(size_t)B * N + (size_t)b * N + n] = wq;
}

// ---------- host launcher ----------
extern "C" void kernel_launch(void* const* d_in, const int* in_sizes, int n_in,
                              void* d_out, int out_size, void* d_ws, size_t ws_size,
                              hipStream_t stream) {
    const float* src = (const float*)d_in[0];
    const float* tgt = (const float*)d_in[1];
    const float* src_n = (const float*)d_in[2];
    const float* tgt_n = (const float*)d_in[3];
    const float* g_p = (const float*)d_in[4];
    const float* k_p = (const float*)d_in[5];
    const float* g_q = (const float*)d_in[6];
    const float* k_q = (const float*)d_in[7];

    int B = in_sizes[4] / 3;          // g_p is (B,3)
    int N = in_sizes[0] / (3 * B);    // src is (B,3,N)
    size_t BN = (size_t)B * N;

    // workspace layout (u64 arrays first for alignment)
    ull* ROWK = (ull*)d_ws;
    ull* COLK = ROWK + BN;
    float* SI = (float*)(COLK + BN);
    float* SNR = SI + 3 * BN;
    float* XX = SNR + 3 * BN;
    float* YY = XX + BN;
    float* NDP = YY + BN;
    float* NDQ = NDP + BN;
    float* TAU = NDQ + BN;  // 2*B floats

    k_prep<<<B, 256, 0, stream>>>(src, tgt, src_n, g_p, g_q,
                                  SI, SNR, XX, YY, COLK, N);

    // 8 waves/block, each wave owns one 16-row tile over all columns
    dim3 gdist((N + 127) / 128, B);
    k_dist<<<gdist, 256, 0, stream>>>(SI, XX, tgt, YY, ROWK, COLK, N);

    k_nn<<<((int)BN + 255) / 256, 256, 0, stream>>>(ROWK, COLK, NDP, NDQ, (int)BN);

    k_median<<<2 * B, 256, N * sizeof(float), stream>>>(NDP, NDQ, TAU, B, N);

    dim3 gw((N + 255) / 256, B);
    k_weights<<<gw, 256, 0, stream>>>(SNR, tgt_n, ROWK, COLK, NDP, NDQ, TAU,
                                      g_q, k_p, k_q, (float*)d_out, B, N);
}